// Mamba3_89343909692086
// MI455X (gfx1250) — compile-verified
//
#include <hip/hip_runtime.h>
#include <hip/hip_bf16.h>
#include <math.h>

// ---------------------------------------------------------------------------
// Problem constants (from the reference)
// ---------------------------------------------------------------------------
#define D_MODEL   1024
#define D_STATE   128
#define HEADDIM   64
#define NHEADS    32
#define D_INNER   2048
#define SEQLEN    1024
#define NANG      32          // SPLIT/2, SPLIT = 64
#define SPLIT     64
#define D_IN_PROJ 4480
// proj split offsets
#define OFF_Z     0
#define OFF_X     2048
#define OFF_B     4096
#define OFF_C     4224
#define OFF_DT    4352
#define OFF_A     4384
#define OFF_TRAP  4416
#define OFF_ANG   4448
#define A_FLOOR   1e-4f
#define EPS_RMS   1e-5f

typedef float v2f __attribute__((ext_vector_type(2)));
typedef float v8f __attribute__((ext_vector_type(8)));

// ---------------------------------------------------------------------------
// CDNA5 async global->LDS helpers (ASYNCcnt-tracked; ISA ch.10 / 15.18.3).
// LDS destination = low 32 bits of the generic pointer (flat LDS aperture).
// ---------------------------------------------------------------------------
__device__ __forceinline__ unsigned lds_off(const void* p) {
  return (unsigned)(unsigned long long)p;
}
__device__ __forceinline__ void async_b32(unsigned dst, const float* src) {
  asm volatile("global_load_async_to_lds_b32 %0, %1, off"
               :: "v"(dst), "v"(src) : "memory");
}
__device__ __forceinline__ void async_b128(unsigned dst, const float* src) {
  asm volatile("global_load_async_to_lds_b128 %0, %1, off"
               :: "v"(dst), "v"(src) : "memory");
}
__device__ __forceinline__ void wait_async0() {
  asm volatile("s_wait_asynccnt 0" ::: "memory");
}

// ---------------------------------------------------------------------------
// FP32 WMMA GEMM:  C[M,N] = A[M,K] * B[K,N], row-major, M,N % 128 == 0,
// K % 16 == 0.  Block = 128 threads = 4 waves; block tile 128x128; each wave
// owns 64x64 = 4x4 grid of 16x16 accumulators (V_WMMA_F32_16X16X4_F32).
// K-chunks of 16 are double-buffered in LDS and streamed in with
// GLOBAL_LOAD_ASYNC_TO_LDS while the previous chunk's 64 WMMAs execute.
// Both tiles are stored K-major (tile[k][m] / tile[k][n], stride 136) so all
// fragment reads are lane-contiguous and the two lane-halves (k vs k+2 rows)
// land on disjoint bank sets.
// ---------------------------------------------------------------------------
#define TILE_MN 128
#define KB      16
#define LDB     136           // LDS row stride (136*4 B: 16B-aligned, +16 bank skew)

__global__ __launch_bounds__(128)
void gemm_f32_wmma(const float* __restrict__ A, const float* __restrict__ B,
                   float* __restrict__ C, int M, int N, int K) {
  __shared__ __align__(16) float As[2][KB * LDB];  // As[k][m]
  __shared__ __align__(16) float Bs[2][KB * LDB];  // Bs[k][n]

  const int tid  = threadIdx.x;
  const int lane = tid & 31;
  const int wave = tid >> 5;
  const int l16  = lane & 15;
  const int half = lane >> 4;          // selects K pair {0,1} vs {2,3}, rows 0-7 vs 8-15
  const int wm   = (wave & 1) * 64;
  const int wn   = (wave >> 1) * 64;
  const int mbase = blockIdx.y * TILE_MN;
  const int nbase = blockIdx.x * TILE_MN;

  // async-stage one 16-deep K-chunk (A: 16 scatter b32/thread, B: 4 b128/thread)
  auto stage = [&](int buf, int kc) {
    const int ka = tid & 15;             // K lane for A
    const int mg = (tid >> 4) * 16;      // M group for A
    #pragma unroll
    for (int i = 0; i < 16; ++i) {
      const int m = mg + i;
      async_b32(lds_off(&As[buf][ka * LDB + m]),
                A + (size_t)(mbase + m) * K + kc + ka);
    }
    #pragma unroll
    for (int i = 0; i < 4; ++i) {
      const int e  = tid + i * 128;
      const int r  = e >> 5;             // K row for B
      const int c4 = (e & 31) * 4;       // N column (float4)
      async_b128(lds_off(&Bs[buf][r * LDB + c4]),
                 B + (size_t)(kc + r) * N + nbase + c4);
    }
  };

  v8f acc[4][4] = {};
  stage(0, 0);
  const int nchunks = K / KB;

  for (int ic = 0; ic < nchunks; ++ic) {
    const int cur = ic & 1;
    wait_async0();                       // this wave's chunk-ic data in LDS
    __syncthreads();                     // -> everyone's data in LDS
    if (ic + 1 < nchunks) stage(cur ^ 1, (ic + 1) * KB);  // overlap with compute

    const float* Asb = As[cur];
    const float* Bsb = Bs[cur];
    #pragma unroll
    for (int k0 = 0; k0 < KB; k0 += 4) {
      const int ka = k0 + 2 * half;
      v2f af[4], bf[4];
      #pragma unroll
      for (int ms = 0; ms < 4; ++ms) {
        const float* p = &Asb[ka * LDB + wm + ms * 16 + l16];
        af[ms][0] = p[0]; af[ms][1] = p[LDB];
      }
      #pragma unroll
      for (int ns = 0; ns < 4; ++ns) {
        const float* p = &Bsb[ka * LDB + wn + ns * 16 + l16];
        bf[ns][0] = p[0]; bf[ns][1] = p[LDB];
      }
      #pragma unroll
      for (int ms = 0; ms < 4; ++ms)
        #pragma unroll
        for (int ns = 0; ns < 4; ++ns)
          acc[ms][ns] = __builtin_amdgcn_wmma_f32_16x16x4_f32(
              false, af[ms], false, bf[ns], (short)0, acc[ms][ns], false, false);
    }
    __syncthreads();                     // compute done before slot reuse
  }

  // store D: VGPR r holds row (r + 8*half), col = l16 within 16x16 tile
  #pragma unroll
  for (int ms = 0; ms < 4; ++ms)
    #pragma unroll
    for (int ns = 0; ns < 4; ++ns)
      #pragma unroll
      for (int r = 0; r < 8; ++r) {
        const int row = mbase + wm + ms * 16 + half * 8 + r;
        const int col = nbase + wn + ns * 16 + l16;
        C[(size_t)row * N + col] = acc[ms][ns][r];
      }
}

// ---------------------------------------------------------------------------
// Per-timestep scalars + rmsnorm of B/C.  grid = SEQLEN, block = 128.
// ---------------------------------------------------------------------------
__device__ __forceinline__ float softplus_f(float x) {
  return x > 20.f ? x : log1pf(expf(x));
}

__global__ __launch_bounds__(128)
void prep_kernel(const float* __restrict__ proj, const float* __restrict__ dt_bias,
                 const float* __restrict__ Bn_w, const float* __restrict__ Cn_w,
                 float* __restrict__ Bn_g, float* __restrict__ Cn_g,
                 float* __restrict__ dt_g, float* __restrict__ a_g,
                 float* __restrict__ cdt_g, float* __restrict__ pdt_g) {
  const int t = blockIdx.x, n = threadIdx.x;
  __shared__ float red[128];
  const float* pr = proj + (size_t)t * D_IN_PROJ;

  float b = pr[OFF_B + n];
  red[n] = b * b; __syncthreads();
  for (int s = 64; s > 0; s >>= 1) { if (n < s) red[n] += red[n + s]; __syncthreads(); }
  const float brms = rsqrtf(red[0] * (1.0f / D_STATE) + EPS_RMS);
  Bn_g[(size_t)t * D_STATE + n] = b * brms * Bn_w[n];
  __syncthreads();

  float c = pr[OFF_C + n];
  red[n] = c * c; __syncthreads();
  for (int s = 64; s > 0; s >>= 1) { if (n < s) red[n] += red[n + s]; __syncthreads(); }
  const float crms = rsqrtf(red[0] * (1.0f / D_STATE) + EPS_RMS);
  Cn_g[(size_t)t * D_STATE + n] = c * crms * Cn_w[n];

  if (n < NHEADS) {
    const int h = n;
    const float dt  = softplus_f(pr[OFF_DT + h] + dt_bias[h]);
    const float Am  = -fmaxf(softplus_f(pr[OFF_A + h]), A_FLOOR);
    const float a   = expf(Am * dt);
    const float lam = 1.f / (1.f + expf(-pr[OFF_TRAP + h]));
    dt_g [t * NHEADS + h] = dt;
    a_g  [t * NHEADS + h] = a;
    cdt_g[t * NHEADS + h] = dt * lam;
    pdt_g[t * NHEADS + h] = dt * (1.f - lam) * a;
  }
}

// ---------------------------------------------------------------------------
// theta[t,h,n] = cumsum_t( ang[t,n] * dt[t,h] ).  One block, 1024 threads.
// ---------------------------------------------------------------------------
__global__ __launch_bounds__(1024)
void theta_kernel(const float* __restrict__ proj, const float* __restrict__ dt_g,
                  float* __restrict__ theta) {
  const int h = threadIdx.x >> 5, n = threadIdx.x & 31;
  float acc = 0.f;
  for (int t = 0; t < SEQLEN; ++t) {
    acc += proj[(size_t)t * D_IN_PROJ + OFF_ANG + n] * dt_g[t * NHEADS + h];
    theta[((size_t)t * NHEADS + h) * NANG + n] = acc;
  }
}

// ---------------------------------------------------------------------------
// Bh/Ch = rope(rmsnorm + head bias).  grid = SEQLEN, block = 128, loop heads.
// ---------------------------------------------------------------------------
__global__ __launch_bounds__(128)
void rope_kernel(const float* __restrict__ Bn_g, const float* __restrict__ Cn_g,
                 const float* __restrict__ B_bias, const float* __restrict__ C_bias,
                 const float* __restrict__ theta,
                 float* __restrict__ Bh, float* __restrict__ Ch) {
  const int t = blockIdx.x, tid = threadIdx.x;
  for (int h = 0; h < NHEADS; ++h) {
    const size_t ob = ((size_t)t * NHEADS + h) * D_STATE;
    if (tid < NANG) {
      const int n = tid;
      const float th = theta[((size_t)t * NHEADS + h) * NANG + n];
      float s, c; __sincosf(th, &s, &c);
      const float b1 = Bn_g[(size_t)t * D_STATE + n]        + B_bias[h * D_STATE + n];
      const float b2 = Bn_g[(size_t)t * D_STATE + n + NANG] + B_bias[h * D_STATE + n + NANG];
      Bh[ob + n]        = b1 * c - b2 * s;
      Bh[ob + n + NANG] = b1 * s + b2 * c;
      const float c1 = Cn_g[(size_t)t * D_STATE + n]        + C_bias[h * D_STATE + n];
      const float c2 = Cn_g[(size_t)t * D_STATE + n + NANG] + C_bias[h * D_STATE + n + NANG];
      Ch[ob + n]        = c1 * c - c2 * s;
      Ch[ob + n + NANG] = c1 * s + c2 * c;
    } else if (tid >= SPLIT) {
      const int n = tid;
      Bh[ob + n] = Bn_g[(size_t)t * D_STATE + n] + B_bias[h * D_STATE + n];
      Ch[ob + n] = Cn_g[(size_t)t * D_STATE + n] + C_bias[h * D_STATE + n];
    }
  }
}

// ---------------------------------------------------------------------------
// Sequential SSM scan.  grid = NHEADS, block = 256.  State h[64][128] lives in
// registers (32 f32/thread).  Step t+1's B/C/x are prefetched into LDS with
// async-to-LDS while step t computes; B/x use a 3-slot ring (the recurrence
// also reads step t-1 as kp/vp), C a 2-slot ring.  Scalars are register-
// prefetched one step ahead.  Epilogue fuses (y + D*x) * silu(z) -> Y.
// ---------------------------------------------------------------------------
__global__ __launch_bounds__(256)
void scan_kernel(const float* __restrict__ proj,
                 const float* __restrict__ Bh, const float* __restrict__ Ch,
                 const float* __restrict__ a_g, const float* __restrict__ cdt_g,
                 const float* __restrict__ pdt_g, const float* __restrict__ D_param,
                 float* __restrict__ Y) {
  const int h   = blockIdx.x;
  const int tid = threadIdx.x;
  const int p   = tid & 63;      // state row (headdim)
  const int ng  = tid >> 6;      // state-col group: n in [ng*32, ng*32+32)

  __shared__ __align__(16) float Bbuf[3][D_STATE];
  __shared__ __align__(16) float Cbuf[2][D_STATE];
  __shared__ __align__(16) float xbuf[3][HEADDIM];
  __shared__ float partial[256];

  // slot 2 holds the t = -1 (zero) state for kp/vp at t = 0
  if (tid < D_STATE) Bbuf[2][tid] = 0.f;
  if (tid < HEADDIM) xbuf[2][tid] = 0.f;

  auto issue = [&](int t) {              // async prefetch of step t inputs
    const int bs = t % 3, cs = t & 1;
    const size_t base = ((size_t)t * NHEADS + h) * D_STATE;
    if (tid < 32) {
      async_b128(lds_off(&Bbuf[bs][tid * 4]), Bh + base + tid * 4);
    } else if (tid < 64) {
      async_b128(lds_off(&Cbuf[cs][(tid - 32) * 4]), Ch + base + (tid - 32) * 4);
    } else if (tid < 80) {
      async_b128(lds_off(&xbuf[bs][(tid - 64) * 4]),
                 proj + (size_t)t * D_IN_PROJ + OFF_X + h * HEADDIM + (tid - 64) * 4);
    }
  };
  issue(0);

  float hreg[32];
  #pragma unroll
  for (int j = 0; j < 32; ++j) hreg[j] = 0.f;
  const float Dp = D_param[h];

  // scalar prefetch registers (t = 0)
  float a_t = a_g[h], cdt = cdt_g[h], pdt = pdt_g[h];

  for (int t = 0; t < SEQLEN; ++t) {
    const int cur = t % 3, prv = (t + 2) % 3, cc = t & 1;
    wait_async0();                       // step-t inputs (this wave's) landed
    __syncthreads();                     // all waves' inputs landed
    if (t + 1 < SEQLEN) issue(t + 1);    // overlap next-step fetch with compute

    float a_n = 0.f, c_n = 0.f, p_n = 0.f;
    if (t + 1 < SEQLEN) {                // register-prefetch next scalars
      a_n = a_g[(t + 1) * NHEADS + h];
      c_n = cdt_g[(t + 1) * NHEADS + h];
      p_n = pdt_g[(t + 1) * NHEADS + h];
    }

    const float cx = cdt * xbuf[cur][p];
    const float pv = pdt * xbuf[prv][p];
    float y = 0.f;
    #pragma unroll
    for (int j = 0; j < 32; ++j) {
      const int n = ng * 32 + j;
      hreg[j] = a_t * hreg[j] + cx * Bbuf[cur][n] + pv * Bbuf[prv][n];
      y += Cbuf[cc][n] * hreg[j];
    }
    partial[tid] = y;
    __syncthreads();

    if (tid < HEADDIM) {
      const float ys = partial[tid] + partial[tid + 64] + partial[tid + 128] + partial[tid + 192];
      const float xv = xbuf[cur][tid];
      const float zv = proj[(size_t)t * D_IN_PROJ + OFF_Z + h * HEADDIM + tid];
      const float sig = 1.f / (1.f + expf(-zv));
      Y[(size_t)t * D_INNER + h * HEADDIM + tid] = (ys + Dp * xv) * (zv * sig);
    }
    a_t = a_n; cdt = c_n; pdt = p_n;
  }
}

// ---------------------------------------------------------------------------
// Launcher
// ---------------------------------------------------------------------------
extern "C" void kernel_launch(void* const* d_in, const int* in_sizes, int n_in,
                              void* d_out, int out_size, void* d_ws, size_t ws_size,
                              hipStream_t stream) {
  const float* u       = (const float*)d_in[0];  // (1,1024,1024)
  const float* W_in    = (const float*)d_in[1];  // (1024,4480)
  const float* dt_bias = (const float*)d_in[2];  // (32,)
  const float* B_bias  = (const float*)d_in[3];  // (32,128)
  const float* C_bias  = (const float*)d_in[4];  // (32,128)
  const float* Bn_w    = (const float*)d_in[5];  // (128,)
  const float* Cn_w    = (const float*)d_in[6];  // (128,)
  const float* D_param = (const float*)d_in[7];  // (32,)
  const float* W_out   = (const float*)d_in[8];  // (2048,1024)
  float* out = (float*)d_out;                    // (1,1024,1024)

  // workspace layout (floats)
  float* ws    = (float*)d_ws;
  float* proj  = ws;                                        // 1024*4480
  float* Bn_g  = proj  + (size_t)SEQLEN * D_IN_PROJ;        // 1024*128
  float* Cn_g  = Bn_g  + (size_t)SEQLEN * D_STATE;          // 1024*128
  float* dt_g  = Cn_g  + (size_t)SEQLEN * D_STATE;          // 1024*32
  float* a_g   = dt_g  + (size_t)SEQLEN * NHEADS;
  float* cdt_g = a_g   + (size_t)SEQLEN * NHEADS;
  float* pdt_g = cdt_g + (size_t)SEQLEN * NHEADS;
  float* theta = pdt_g + (size_t)SEQLEN * NHEADS;           // 1024*32*32
  float* BhA   = theta + (size_t)SEQLEN * NHEADS * NANG;    // 1024*32*128
  float* ChA   = BhA   + (size_t)SEQLEN * NHEADS * D_STATE;
  float* Yact  = ChA   + (size_t)SEQLEN * NHEADS * D_STATE; // 1024*2048

  // 1) in-projection GEMM: proj = u @ W_in   (1024 x 4480 x K=1024)
  gemm_f32_wmma<<<dim3(D_IN_PROJ / 128, SEQLEN / 128), 128, 0, stream>>>(
      u, W_in, proj, SEQLEN, D_IN_PROJ, D_MODEL);

  // 2) per-step scalars + rmsnorm
  prep_kernel<<<SEQLEN, 128, 0, stream>>>(proj, dt_bias, Bn_w, Cn_w,
                                          Bn_g, Cn_g, dt_g, a_g, cdt_g, pdt_g);

  // 3) theta cumulative sum over time
  theta_kernel<<<1, 1024, 0, stream>>>(proj, dt_g, theta);

  // 4) bias + rope -> Bh, Ch
  rope_kernel<<<SEQLEN, 128, 0, stream>>>(Bn_g, Cn_g, B_bias, C_bias, theta, BhA, ChA);

  // 5) sequential scan with async next-step prefetch -> Yact
  scan_kernel<<<NHEADS, 256, 0, stream>>>(proj, BhA, ChA, a_g, cdt_g, pdt_g,
                                          D_param, Yact);

  // 6) out-projection GEMM: out = Yact @ W_out   (1024 x 1024 x K=2048)
  gemm_f32_wmma<<<dim3(D_MODEL / 128, SEQLEN / 128), 128, 0, stream>>>(
      Yact, W_out, out, SEQLEN, D_MODEL, D_INNER);
}